// BlinnPhongModel_43181601194532
// MI455X (gfx1250) — compile-verified
//
#include <hip/hip_runtime.h>
#include <hip/hip_bf16.h>

#define BP_EPS 1e-12f

// ---------------------------------------------------------------------------
// Main kernel: 4 Blinn-Phong points per thread, all global traffic as b128.
//   in : (N,3,3) f32, rows = light / normal / view  (36 B per point)
//   out: (N,3)   f32                                 (12 B per point)
// Memory-bound: 192 MB total -> ~8.3 us floor at 23.3 TB/s.
// ---------------------------------------------------------------------------
__global__ __launch_bounds__(256) void blinn_phong_q4_kernel(
    const float4* __restrict__ in4,   // nquad * 9 float4
    const float*  __restrict__ kd,    // 3
    const float*  __restrict__ ks,    // 3
    const float*  __restrict__ pp,    // 1
    float4*       __restrict__ out4,  // nquad * 3 float4
    int nquad)
{
    int idx = blockIdx.x * blockDim.x + threadIdx.x;
    if (idx >= nquad) return;

    const float kd0 = kd[0], kd1 = kd[1], kd2 = kd[2];
    const float ks0 = ks[0], ks1 = ks[1], ks2 = ks[2];
    const float p   = pp[0];

    // Nine coalesced-in-aggregate b128 loads: 144 contiguous bytes = 4 points.
    float4 r[9];
    const float4* base = in4 + (size_t)idx * 9;
#pragma unroll
    for (int k = 0; k < 9; ++k) r[k] = base[k];

    // Unpack to scalar regs (constant indices -> full SROA, no scratch).
    float f[36];
#pragma unroll
    for (int k = 0; k < 9; ++k) {
        f[4 * k + 0] = r[k].x;
        f[4 * k + 1] = r[k].y;
        f[4 * k + 2] = r[k].z;
        f[4 * k + 3] = r[k].w;
    }

    float o[12];
#pragma unroll
    for (int j = 0; j < 4; ++j) {
        const float lx = f[j * 9 + 0], ly = f[j * 9 + 1], lz = f[j * 9 + 2];
        const float nx = f[j * 9 + 3], ny = f[j * 9 + 4], nz = f[j * 9 + 5];
        const float vx = f[j * 9 + 6], vy = f[j * 9 + 7], vz = f[j * 9 + 8];

        // Diffuse term: ln = max(0, L.N)
        const float ln = fmaxf(0.0f, lx * nx + ly * ny + lz * nz);

        // Half vector h = L + V; nh = max(0, N.h / max(||h||, eps))
        const float hx = lx + vx, hy = ly + vy, hz = lz + vz;
        const float hn = fmaxf(sqrtf(hx * hx + hy * hy + hz * hz), BP_EPS);
        const float nh = fmaxf(0.0f, __fdividef(nx * hx + ny * hy + nz * hz, hn));

        // Specular: nh^p via v_log2/v_exp2 (nh==0 -> exp2(-inf) == 0, matches ref)
        const float spec = __powf(nh, p);

        o[j * 3 + 0] = ks0 * spec + kd0 * ln;
        o[j * 3 + 1] = ks1 * spec + kd1 * ln;
        o[j * 3 + 2] = ks2 * spec + kd2 * ln;
    }

    // Three aligned b128 stores: 48 contiguous bytes = 4 points of RGB.
    float4* ob = out4 + (size_t)idx * 3;
    ob[0] = make_float4(o[0], o[1], o[2],  o[3]);
    ob[1] = make_float4(o[4], o[5], o[6],  o[7]);
    ob[2] = make_float4(o[8], o[9], o[10], o[11]);
}

// ---------------------------------------------------------------------------
// TDM probe kernel (not launched): establishes which tensor_load_to_lds arity
// this toolchain accepts and that s_wait_tensorcnt lowers. Descriptors are
// zero-filled (probe-verified pattern) — next round can build real D# groups
// for an LDS-staged variant if the instruction path is confirmed.
// ---------------------------------------------------------------------------
#if defined(__gfx1250__) && __has_builtin(__builtin_amdgcn_tensor_load_to_lds)
typedef unsigned int bp_u32x4 __attribute__((ext_vector_type(4)));
typedef int          bp_i32x8 __attribute__((ext_vector_type(8)));
typedef int          bp_i32x4 __attribute__((ext_vector_type(4)));

__global__ void blinn_phong_tdm_probe_kernel()
{
    bp_u32x4 g0 = (bp_u32x4)0u;
    bp_i32x8 g1 = (bp_i32x8)0;
    bp_i32x4 g2 = (bp_i32x4)0;
    bp_i32x4 g3 = (bp_i32x4)0;
#if __has_include(<hip/amd_detail/amd_gfx1250_TDM.h>)
    // amdgpu-toolchain (clang-23 / therock-10.0 headers): 6-arg form
    bp_i32x8 g4 = (bp_i32x8)0;
    __builtin_amdgcn_tensor_load_to_lds(g0, g1, g2, g3, g4, 0);
#else
    // ROCm 7.2 (clang-22): 5-arg form
    __builtin_amdgcn_tensor_load_to_lds(g0, g1, g2, g3, 0);
#endif
#if __has_builtin(__builtin_amdgcn_s_wait_tensorcnt)
    __builtin_amdgcn_s_wait_tensorcnt(0);
#endif
}
#endif

// ---------------------------------------------------------------------------
// Host-side launcher. Inputs (setup_inputs order): inputs, kd, ks, p.
// ---------------------------------------------------------------------------
extern "C" void kernel_launch(void* const* d_in, const int* in_sizes, int n_in,
                              void* d_out, int out_size, void* d_ws, size_t ws_size,
                              hipStream_t stream)
{
    const float4* in4 = (const float4*)d_in[0];
    const float*  kd  = (const float*)d_in[1];
    const float*  ks  = (const float*)d_in[2];
    const float*  p   = (const float*)d_in[3];
    float4*       out = (float4*)d_out;

    const int npoints = in_sizes[0] / 9;   // N
    const int nquad   = npoints / 4;       // 4 points per thread (N is a multiple of 4)

    const int block = 256;
    const int grid  = (nquad + block - 1) / block;
    blinn_phong_q4_kernel<<<grid, block, 0, stream>>>(in4, kd, ks, p, out, nquad);
}